// YOLOLayer_69114613728950
// MI455X (gfx1250) — compile-verified
//
#include <hip/hip_runtime.h>
#include <hip/hip_bf16.h>

// ---------------------------------------------------------------------------
// YOLO detection head for MI455X (gfx1250, wave32).
// Memory-bound elementwise transform + planar->interleaved transpose.
// 188.5 MB total traffic @ 23.3 TB/s  => ~8.1 us roofline.
// No matmul structure => WMMA not applicable; optimize HBM streaming + TRANS.
// Streaming data (read-once / write-once) => nontemporal TH policy on all
// global traffic so the 192MB L2 / WGP$ aren't polluted with dead lines.
// Sigmoid via v_tanh_f32 (confirmed available): 1 TRANS op per activation.
// ---------------------------------------------------------------------------

#define NUM_CLASSES 80
#define NCP5        85          // NUM_CLASSES + 5
#define NUM_ANCHORS 3
#define INP_DIM     608
#define GRID        76
#define GG          (GRID * GRID)   // 5776
#define BATCH       16
#define TPB         128             // 4 wave32 per block
#define BPP         ((GG + TPB - 1) / TPB)  // blocks per (b, a) plane = 46

// true clang vector type: required by __builtin_nontemporal_store
typedef __attribute__((ext_vector_type(4))) float v4f;

#if __has_builtin(__builtin_amdgcn_tanhf)
#pragma message("CDNA5 path: sigmoid via v_tanh_f32 (1 TRANS op)")
#else
#pragma message("Fallback path: sigmoid via v_exp_f32 + v_rcp_f32 (2 TRANS ops)")
#endif

__device__ __forceinline__ float fast_exp(float x) {
#if __has_builtin(__builtin_amdgcn_exp2f)
    return __builtin_amdgcn_exp2f(x * 1.4426950408889634f);  // v_exp_f32
#else
    return __expf(x);
#endif
}

__device__ __forceinline__ float fast_rcp(float x) {
#if __has_builtin(__builtin_amdgcn_rcpf)
    return __builtin_amdgcn_rcpf(x);                          // v_rcp_f32
#else
    return 1.0f / x;
#endif
}

__device__ __forceinline__ float fast_sigmoid(float x) {
#if __has_builtin(__builtin_amdgcn_tanhf)
    // 1 TRANS op instead of 2: sigmoid(x) = 0.5*tanh(x/2) + 0.5  (v_tanh_f32)
    return __builtin_fmaf(0.5f, __builtin_amdgcn_tanhf(0.5f * x), 0.5f);
#else
    return fast_rcp(1.0f + fast_exp(-x));
#endif
}

__global__ __launch_bounds__(TPB) void yolo_head_kernel(
    const float* __restrict__ pred,      // [B, A*85, G, G]
    const float* __restrict__ anchors,   // [A, 2]
    float* __restrict__ out)             // [B, A*G*G, 85]
{
    // LDS tile: 128 positions x 85 channels, stride 85 (gcd(85,64)=1 => no bank conflicts)
    __shared__ float tile[TPB * NCP5];   // 43,520 bytes

    const int tid   = threadIdx.x;
    const int blk   = blockIdx.x;
    const int plane = blk / BPP;         // b*A + a
    const int chunk = blk % BPP;
    const int a     = plane % NUM_ANCHORS;
    const int pos0  = chunk * TPB;

    const int pos = pos0 + tid;
    const int cp  = pos < GG ? pos : (GG - 1);   // clamp: keep loads in-bounds, copy-out masks invalids
    const int gy  = cp / GRID;
    const int gx  = cp - gy * GRID;

    const float stride = (float)INP_DIM / (float)GRID;   // 8.0
    const float aw_s   = anchors[2 * a + 0] / stride;
    const float ah_s   = anchors[2 * a + 1] / stride;
    const float gxf    = (float)gx;
    const float gyf    = (float)gy;

    const float* __restrict__ plane_base = pred + (size_t)plane * NCP5 * GG;

    // Speculative prefetch of the next chunk's cachelines -> global_prefetch_b8
    if (pos0 + TPB < GG) {
        const float* pf = plane_base + (pos0 + TPB + tid);
#pragma unroll
        for (int c = 0; c < NCP5; c += 16)
            __builtin_prefetch(pf + (size_t)c * GG, 0, 0);
    }

    // ---- load (coalesced: 32 consecutive floats per wave per channel,
    //      nontemporal: read-once streaming), transform, stage into LDS ----
    float* __restrict__ row = &tile[tid * NCP5];
#pragma unroll
    for (int c = 0; c < NCP5; ++c) {
        const float v = __builtin_nontemporal_load(plane_base + (size_t)c * GG + cp);
        float r;
        if      (c == 0) r = (fast_sigmoid(v) + gxf) * stride;        // bx
        else if (c == 1) r = (fast_sigmoid(v) + gyf) * stride;        // by
        else if (c == 2) r = fast_exp(v) * aw_s * stride;             // bw
        else if (c == 3) r = fast_exp(v) * ah_s * stride;             // bh
        else             r = fast_sigmoid(v);                         // conf + classes
        row[c] = r;
    }

    __syncthreads();   // s_barrier_signal/_wait + s_wait_dscnt

    // ---- cooperative coalesced copy-out: LDS tile -> contiguous global ---
    const int nvalid = (GG - pos0) < TPB ? (GG - pos0) : TPB;
    const int total  = nvalid * NCP5;
    const size_t out_base = (size_t)(plane * GG + pos0) * NCP5;

    // alignment fix-up so the body uses global_store_b128
    int head = (int)((4 - (out_base & 3)) & 3);
    if (head > total) head = total;
    if (tid < head)
        __builtin_nontemporal_store(tile[tid], out + out_base + tid);

    const int nvec = (total - head) >> 2;
    v4f* __restrict__ outv = (v4f*)(out + out_base + head);
    for (int j = tid; j < nvec; j += TPB) {
        const int idx = head + (j << 2);
        v4f vv;
        vv.x = tile[idx + 0];
        vv.y = tile[idx + 1];
        vv.z = tile[idx + 2];
        vv.w = tile[idx + 3];
        __builtin_nontemporal_store(vv, outv + j);   // global_store_b128 th:NT
    }

    const int done = head + (nvec << 2);
    const int rem  = total - done;
    if (tid < rem)
        __builtin_nontemporal_store(tile[done + tid], out + out_base + done + tid);
}

extern "C" void kernel_launch(void* const* d_in, const int* in_sizes, int n_in,
                              void* d_out, int out_size, void* d_ws, size_t ws_size,
                              hipStream_t stream) {
    (void)in_sizes; (void)n_in; (void)d_ws; (void)ws_size; (void)out_size;
    const float* pred    = (const float*)d_in[0];
    const float* anchors = (const float*)d_in[1];
    float* out           = (float*)d_out;

    const int nblocks = BATCH * NUM_ANCHORS * BPP;   // 16*3*46 = 2208
    yolo_head_kernel<<<nblocks, TPB, 0, stream>>>(pred, anchors, out);
}